// Phi_Layer_multipleT_32212254720398
// MI455X (gfx1250) — compile-verified
//
#include <hip/hip_runtime.h>
#include <hip/hip_bf16.h>

// B=32, Nt=32, Nr=32, Mt=16, Mr=16, G=64 (G^2=4096), NRF=4, S=32
#define GSQ    4096
#define NT     32
#define NR     32
#define MT     16
#define MR     16
#define BATCH  32
#define SC     32
#define INV_SQRT_NT 0.17677669529663687f  // 1/sqrt(32)

typedef __attribute__((ext_vector_type(2))) float v2f;
typedef __attribute__((ext_vector_type(8))) float v8f;

// D = A(16x4, f32) * B(4x16, f32) + C(16x16, f32)
static __device__ inline v8f wmma4(v2f a, v2f b, v8f c) {
    return __builtin_amdgcn_wmma_f32_16x16x4_f32(
        /*neg_a=*/false, a, /*neg_b=*/false, b,
        /*c_mod=*/(short)0, c, /*reuse_a=*/false, /*reuse_b=*/false);
}

// ---------------------------------------------------------------------------
// Kernel 1: T2[(mt*32+nr), g] = sum_nt F^T[mt,nt] * K2[(nt*32+nr), g]
//   F = (cos + i sin)(kernel_F)/sqrt(Nt).  Complex GEMM M=16(mt) K=32(nt)
//   N=4096(g) per nr.  Batch-independent: done once into d_ws, interleaved
//   float2 {Re,Im} so consumers get both with one b64 load.
//   grid = (8 g-blocks of 512, 32 nr), block = 256 (8 waves x 4 tiles x 16)
// ---------------------------------------------------------------------------
__global__ __launch_bounds__(256) void t2_kernel(
    const float*  __restrict__ kF,   // [Nt, Mt]
    const float2* __restrict__ K2,   // [Nt*Nr, G^2] {re,im}
    float2*       __restrict__ T2)   // [Mt*Nr, G^2] {re,im}
{
    const int nr   = blockIdx.y;
    const int lane = threadIdx.x & 31;
    const int wave = threadIdx.x >> 5;
    const int m    = lane & 15;      // mt row (A) / g column (B,C)
    const int hh   = lane >> 4;      // lane-half
    const int g0w  = blockIdx.x * 512 + wave * 64;

    // Per-lane bases; all loop offsets below are compile-time constants.
    const float2* __restrict__ Kb =
        K2 + ((size_t)(2 * hh * NR + nr)) * GSQ + g0w + m;       // nt = 2hh (+k)
    float2* __restrict__ Ob =
        T2 + ((size_t)(8 * hh * NR + nr)) * GSQ + g0w + m;       // mt = 8hh (+i)

    v8f cr[4], ci[4];
    const v8f z = {};
#pragma unroll
    for (int t = 0; t < 4; ++t) { cr[t] = z; ci[t] = z; }

#pragma unroll
    for (int kc = 0; kc < 8; ++kc) {          // nt in chunks of 4
        v2f aR, aI, aIn;
#pragma unroll
        for (int v = 0; v < 2; ++v) {
            const int nt = kc * 4 + v + 2 * hh;
            const float ph = kF[nt * MT + m];
            const float c = __cosf(ph) * INV_SQRT_NT;
            const float s = __sinf(ph) * INV_SQRT_NT;
            aR[v] = c; aI[v] = s; aIn[v] = -s;
        }
#pragma unroll
        for (int t = 0; t < 4; ++t) {
            v2f bX, bY;
#pragma unroll
            for (int v = 0; v < 2; ++v) {
                const float2 xy = Kb[(size_t)((kc * 4 + v) * NR) * GSQ + t * 16];
                bX[v] = xy.x;
                bY[v] = xy.y;
            }
            // (FR + iFI)(X + iY): Re = FR@X - FI@Y ; Im = FR@Y + FI@X
            cr[t] = wmma4(aR,  bX, cr[t]);
            cr[t] = wmma4(aIn, bY, cr[t]);
            ci[t] = wmma4(aR,  bY, ci[t]);
            ci[t] = wmma4(aI,  bX, ci[t]);
        }
    }

#pragma unroll
    for (int t = 0; t < 4; ++t)
#pragma unroll
        for (int i = 0; i < 8; ++i)
            Ob[(size_t)(i * NR) * GSQ + t * 16] = make_float2(cr[t][i], ci[t][i]);
}

// ---------------------------------------------------------------------------
// Kernel 2: Phi[b, mt*16+mr, g] = sum_nr conj(W[b,nr,mr]) * T2[(mt,nr), g]
//   Complex GEMM M=16(mr) K=32(nr) N=4096(g) per (b,mt). The 8.6-GFLOP hot
//   loop; Phi write (256 MB) is the HBM bound. T2 (16.8 MB) is L2-resident.
//   grid = (8 g-blocks, 16 mt, 32 b), block = 256
// ---------------------------------------------------------------------------
__global__ __launch_bounds__(256) void phi_kernel(
    const float2* __restrict__ W,    // [B, Nr, Mr] {re,im}
    const float2* __restrict__ T2,   // [Mt*Nr, G^2] {re,im}
    float2*       __restrict__ phi)  // [B, Mt*Mr, G^2] {re,im}
{
    const int b    = blockIdx.z;
    const int mt   = blockIdx.y;
    const int lane = threadIdx.x & 31;
    const int wave = threadIdx.x >> 5;
    const int m    = lane & 15;
    const int hh   = lane >> 4;
    const int g0w  = blockIdx.x * 512 + wave * 64;

    const float2* __restrict__ Wb =
        W + (size_t)(b * NR + 2 * hh) * MR + m;                  // nr = 2hh (+k)
    const float2* __restrict__ Bb =
        T2 + ((size_t)(mt * NR + 2 * hh)) * GSQ + g0w + m;
    float2* __restrict__ Ob =
        phi + ((size_t)(b * (MT * MR) + mt * MR + 8 * hh)) * GSQ + g0w + m;

    v8f cr[4], ci[4];
    const v8f z = {};
#pragma unroll
    for (int t = 0; t < 4; ++t) { cr[t] = z; ci[t] = z; }

#pragma unroll
    for (int kc = 0; kc < 8; ++kc) {          // nr in chunks of 4
        v2f aP, aQ, aQn;                       // W^H = P - iQ, P=Wr^T, Q=Wi^T
#pragma unroll
        for (int v = 0; v < 2; ++v) {
            const float2 w = Wb[(kc * 4 + v) * MR];
            aP[v]  = w.x;
            aQ[v]  = w.y;
            aQn[v] = -w.y;
        }
#pragma unroll
        for (int t = 0; t < 4; ++t) {
            v2f bX, bY;
#pragma unroll
            for (int v = 0; v < 2; ++v) {
                const float2 xy = Bb[(size_t)((kc * 4 + v)) * GSQ + t * 16];
                bX[v] = xy.x;
                bY[v] = xy.y;
            }
            // (P - iQ)(X + iY): Re = P@X + Q@Y ; Im = P@Y - Q@X
            cr[t] = wmma4(aP,  bX, cr[t]);
            cr[t] = wmma4(aQ,  bY, cr[t]);
            ci[t] = wmma4(aP,  bY, ci[t]);
            ci[t] = wmma4(aQn, bX, ci[t]);
        }
    }

#pragma unroll
    for (int t = 0; t < 4; ++t)
#pragma unroll
        for (int i = 0; i < 8; ++i)
            Ob[(size_t)i * GSQ + t * 16] = make_float2(cr[t][i], ci[t][i]);
}

// ---------------------------------------------------------------------------
// Kernel 3: y[b, mt*16+mr, s] = sum_nt F^T[mt,nt]*U[nt,s] + eff[mt,s]
//   U[nt,s]  = sum_nr conj(W[b,nr,mr]) * H[b,nr,nt,s]
//   eff[mt,s]= sum_nr conj(W[b,nr,mr]) * noise[b, mr/4, nr, mt*32+s]
//   ~0.3 GFLOP total: plain VALU + LDS staging. grid=(mr,b), block=256.
// ---------------------------------------------------------------------------
__global__ __launch_bounds__(256) void y_kernel(
    const float* __restrict__ W,      // [B, Nr, Mr, 2]
    const float* __restrict__ noise,  // [B, 4, Nr, Mt*S, 2]
    const float* __restrict__ H,      // [B, Nr, Nt, S, 2]
    const float* __restrict__ kF,     // [Nt, Mt]
    float* __restrict__ y)            // [B, Mt*Mr, S, 2]
{
    __shared__ float UR[NT][SC], UI[NT][SC];
    __shared__ float ER[MT][SC], EI[MT][SC];
    __shared__ float Wr[NR], Wi[NR];

    const int mr  = blockIdx.x;
    const int b   = blockIdx.y;
    const int tid = threadIdx.x;

    if (tid < NR) {
        const float* wp = W + (((size_t)(b * NR + tid)) * MR + mr) * 2;
        Wr[tid] = wp[0];
        Wi[tid] = wp[1];
    }
    __syncthreads();

    // U[nt][s]
    for (int e = tid; e < NT * SC; e += 256) {
        const int nt = e >> 5, s = e & 31;
        float ar = 0.f, ai = 0.f;
        for (int nr = 0; nr < NR; ++nr) {
            const float* hp = H + ((((size_t)(b * NR + nr)) * NT + nt) * SC + s) * 2;
            const float hr = hp[0], hi = hp[1];
            ar += Wr[nr] * hr + Wi[nr] * hi;   // conj(W)*H
            ai += Wr[nr] * hi - Wi[nr] * hr;
        }
        UR[nt][s] = ar; UI[nt][s] = ai;
    }

    // eff[mt][s]
    const int r = mr >> 2;
    for (int e = tid; e < MT * SC; e += 256) {
        const int mt = e >> 5, s = e & 31;
        const int x = mt * SC + s;
        float ar = 0.f, ai = 0.f;
        for (int nr = 0; nr < NR; ++nr) {
            const float* np = noise + ((((size_t)(b * 4 + r)) * NR + nr) * (MT * SC) + x) * 2;
            const float vr = np[0], vi = np[1];
            ar += Wr[nr] * vr + Wi[nr] * vi;
            ai += Wr[nr] * vi - Wi[nr] * vr;
        }
        ER[mt][s] = ar; EI[mt][s] = ai;
    }
    __syncthreads();

    for (int e = tid; e < MT * SC; e += 256) {
        const int mt = e >> 5, s = e & 31;
        float ar = ER[mt][s], ai = EI[mt][s];
        for (int nt = 0; nt < NT; ++nt) {
            const float ph = kF[nt * MT + mt];
            const float fr = __cosf(ph) * INV_SQRT_NT;
            const float fi = __sinf(ph) * INV_SQRT_NT;
            ar += fr * UR[nt][s] - fi * UI[nt][s];
            ai += fr * UI[nt][s] + fi * UR[nt][s];
        }
        const size_t o = (((size_t)(b * (MT * MR)) + mt * MR + mr) * SC + s) * 2;
        y[o]     = ar;
        y[o + 1] = ai;
    }
}

// ---------------------------------------------------------------------------
extern "C" void kernel_launch(void* const* d_in, const int* in_sizes, int n_in,
                              void* d_out, int out_size, void* d_ws, size_t ws_size,
                              hipStream_t stream) {
    (void)in_sizes; (void)n_in; (void)out_size; (void)ws_size;
    const float* W     = (const float*)d_in[0];  // [32,32,16,2]
    const float* noise = (const float*)d_in[1];  // [32,4,32,512,2]
    const float* H     = (const float*)d_in[2];  // [32,32,32,32,2]
    const float* kF    = (const float*)d_in[3];  // [32,16]
    const float* K2    = (const float*)d_in[4];  // [1024,4096,2]

    float* out  = (float*)d_out;
    float* yout = out + (size_t)BATCH * (MT * MR) * GSQ * 2;   // after Phi_ri

    float2* T2 = (float2*)d_ws;                                // 16.8 MB interleaved

    t2_kernel <<<dim3(GSQ / 512, NR, 1),     256, 0, stream>>>(kF, (const float2*)K2, T2);
    phi_kernel<<<dim3(GSQ / 512, MT, BATCH), 256, 0, stream>>>((const float2*)W, T2, (float2*)out);
    y_kernel  <<<dim3(MR, BATCH, 1),         256, 0, stream>>>(W, noise, H, kF, yout);
}